// PtrNet1_6270652252361
// MI455X (gfx1250) — compile-verified
//
#include <hip/hip_runtime.h>
#include <hip/hip_bf16.h>
#include <math.h>

typedef __attribute__((ext_vector_type(2))) float v2f;
typedef __attribute__((ext_vector_type(8))) float v8f;

#define B_    128
#define N_    2000
#define D_    256
#define H_    128
#define M_    4
#define CH_   16          // flash chunks per batch row
#define NCH_  (N_ / CH_)  // 125 keys per chunk
#define NEG_  (-1.0e8f)
#define CLIP_ 10.0f
#define SCALE_ 0.0625f    // 1/sqrt(256)

__device__ __forceinline__ float elu1(float x) { return x > 0.f ? x : (expf(x) - 1.f); }

// ---------------- 1. ExEmbedding + initial query [v1 | ex] ----------------
__global__ void k_embed(const float* __restrict__ rs,
                        const float* __restrict__ E1, const float* __restrict__ b1,
                        const float* __restrict__ E2, const float* __restrict__ b2,
                        const float* __restrict__ E3, const float* __restrict__ b3,
                        const float* __restrict__ v1, float* __restrict__ query) {
  int b = threadIdx.x;
  if (b >= B_) return;
  float r[6];
  for (int i = 0; i < 6; ++i) r[i] = rs[b * 6 + i];
  float h2[64];
  for (int k = 0; k < 64; ++k) h2[k] = b2[k];
  for (int j = 0; j < 84; ++j) {
    float s = b1[j];
    for (int i = 0; i < 6; ++i) s += r[i] * E1[i * 84 + j];
    s = elu1(s);
    for (int k = 0; k < 64; ++k) h2[k] += s * E2[j * 64 + k];
  }
  for (int k = 0; k < 64; ++k) h2[k] = elu1(h2[k]);
  for (int e = 0; e < H_; ++e) {
    float s = b3[e];
    for (int k = 0; k < 64; ++k) s += h2[k] * E3[k * H_ + e];
    query[b * D_ + H_ + e] = s;      // ex part
    query[b * D_ + e]      = v1[e];  // h_pi part
  }
}

// ---------------- 2a. qh[b,m,e] = sum_d query[b,d] * Wq[m,d,e] ----------------
__global__ void k_qh(const float* __restrict__ query, const float* __restrict__ Wq,
                     float* __restrict__ qh) {
  int idx = blockIdx.x * 256 + threadIdx.x;  // B*M*D = 131072
  int e = idx & (D_ - 1), m = (idx >> 8) & 3, b = idx >> 10;
  const float* q = query + b * D_;
  const float* w = Wq + m * D_ * D_ + e;
  float s = 0.f;
  for (int d = 0; d < D_; ++d) s += q[d] * w[d * D_];
  qh[idx] = s;
}

// ---------------- 2b. qk[b,m,d] = scale * sum_e qh[b,m,e] * Wk[m,d,e] ----------------
__global__ void k_qk(const float* __restrict__ qh, const float* __restrict__ Wk,
                     float* __restrict__ qk) {
  int idx = blockIdx.x * 256 + threadIdx.x;
  int d = idx & (D_ - 1), m = (idx >> 8) & 3, b = idx >> 10;
  const float* q = qh + (b * M_ + m) * D_;
  const float* w = Wk + m * D_ * D_ + d * D_;
  float s = 0.f;
  for (int e = 0; e < D_; ++e) s += q[e] * w[e];
  qk[idx] = SCALE_ * s;
}

// ---------------- 3. fused flash glimpse: scores + online softmax + weighted sum ----
// one wave per (b, chunk); acc[4][256] distributed across lanes (8 f32/lane/head).
__global__ __launch_bounds__(256) void k_glimpse_flash(
    const float* __restrict__ F, const float* __restrict__ qk,
    const int* __restrict__ mask,
    float* __restrict__ Mp, float* __restrict__ Sp, float* __restrict__ accp) {
  int lane = threadIdx.x & 31;
  int wave = threadIdx.x >> 5;
  int unit = blockIdx.x * 8 + wave;  // 0..2047
  int b = unit >> 4;
  int ch = unit & (CH_ - 1);
  int n0 = ch * NCH_, n1 = n0 + NCH_;

  // per-lane slice of qk (scale already folded in): d = lane + 32*j
  float qr[M_][8];
#pragma unroll
  for (int m = 0; m < M_; ++m)
#pragma unroll
    for (int j = 0; j < 8; ++j)
      qr[m][j] = qk[(b * M_ + m) * D_ + lane + 32 * j];

  float Mx[M_], S[M_], acc[M_][8];
#pragma unroll
  for (int m = 0; m < M_; ++m) {
    Mx[m] = -INFINITY; S[m] = 0.f;
#pragma unroll
    for (int j = 0; j < 8; ++j) acc[m][j] = 0.f;
  }

  for (int n = n0; n < n1; ++n) {
    const float* fr = F + ((long)b * N_ + n) * D_;
    float f[8];
#pragma unroll
    for (int j = 0; j < 8; ++j) f[j] = fr[lane + 32 * j];  // 8 coalesced 128B loads
    int mk = mask[b * N_ + n];
    float s[M_];
#pragma unroll
    for (int m = 0; m < M_; ++m) {
      float p = 0.f;
#pragma unroll
      for (int j = 0; j < 8; ++j) p += f[j] * qr[m][j];
#pragma unroll
      for (int o = 1; o < 32; o <<= 1) p += __shfl_xor(p, o, 32);
      s[m] = (mk > 0) ? p : NEG_;
    }
#pragma unroll
    for (int m = 0; m < M_; ++m) {
      float nm = fmaxf(Mx[m], s[m]);        // nm >= NEG_ > -inf always
      float sc = expf(Mx[m] - nm);          // first iter: exp(-inf)=0
      float w  = expf(s[m] - nm);
      S[m] = S[m] * sc + w;
#pragma unroll
      for (int j = 0; j < 8; ++j) acc[m][j] = acc[m][j] * sc + w * f[j];
      Mx[m] = nm;
    }
  }

  if (lane == 0) {
#pragma unroll
    for (int m = 0; m < M_; ++m) {
      Mp[(b * CH_ + ch) * M_ + m] = Mx[m];
      Sp[(b * CH_ + ch) * M_ + m] = S[m];
    }
  }
#pragma unroll
  for (int m = 0; m < M_; ++m)
#pragma unroll
    for (int j = 0; j < 8; ++j)
      accp[((long)((b * CH_ + ch) * M_ + m)) * D_ + lane + 32 * j] = acc[m][j];
}

// ---------------- 4. merge flash partials -> fbar[b,m,d] ----------------
__global__ void k_glimpse_merge(const float* __restrict__ Mp, const float* __restrict__ Sp,
                                const float* __restrict__ accp, float* __restrict__ fbar) {
  int bm = blockIdx.x;  // b*4 + m
  int b = bm >> 2, m = bm & 3;
  int d = threadIdx.x;
  float mx = -INFINITY;
  for (int ch = 0; ch < CH_; ++ch) mx = fmaxf(mx, Mp[(b * CH_ + ch) * M_ + m]);
  float stot = 0.f, a = 0.f;
  for (int ch = 0; ch < CH_; ++ch) {
    float e = expf(Mp[(b * CH_ + ch) * M_ + m] - mx);
    stot += Sp[(b * CH_ + ch) * M_ + m] * e;
    a += accp[((long)((b * CH_ + ch) * M_ + m)) * D_ + d] * e;
  }
  fbar[(b * M_ + m) * D_ + d] = a / stot;
}

// ---------------- 5a. g[b,m,e] = sum_d fbar[b,m,d] * Wv[m,d,e] ----------------
__global__ void k_g(const float* __restrict__ fbar, const float* __restrict__ Wv,
                    float* __restrict__ g) {
  int idx = blockIdx.x * 256 + threadIdx.x;
  int e = idx & (D_ - 1), m = (idx >> 8) & 3, b = idx >> 10;
  const float* fb = fbar + (b * M_ + m) * D_;
  const float* w = Wv + m * D_ * D_ + e;
  float s = 0.f;
  for (int d = 0; d < D_; ++d) s += fb[d] * w[d * D_];
  g[idx] = s;
}

// ---------------- 5b. query2[b,d] = sum_m sum_e g[b,m,e]*Wo[m,e,d]; also 2nd output ----
__global__ void k_query2(const float* __restrict__ g, const float* __restrict__ Wo,
                         float* __restrict__ query2, float* __restrict__ out) {
  int idx = blockIdx.x * 256 + threadIdx.x;  // B*D
  int d = idx & (D_ - 1), b = idx >> 8;
  float s = 0.f;
  for (int m = 0; m < M_; ++m) {
    const float* gg = g + (b * M_ + m) * D_;
    const float* w = Wo + m * D_ * D_ + d;
    for (int e = 0; e < D_; ++e) s += gg[e] * w[e * D_];
  }
  query2[idx] = s;
  out[(long)B_ * N_ + idx] = s;
}

// ---------------- 5c. q2 = query2 @ Wq2 ----------------
__global__ void k_q2(const float* __restrict__ query2, const float* __restrict__ Wq2,
                     float* __restrict__ q2) {
  int idx = blockIdx.x * 256 + threadIdx.x;
  int e = idx & (D_ - 1), b = idx >> 8;
  const float* q = query2 + b * D_;
  float s = 0.f;
  for (int d = 0; d < D_; ++d) s += q[d] * Wq2[d * D_ + e];
  q2[idx] = s;
}

// ---------------- pack Wref2 into per-lane B-fragment layout ----------------
// frag(ks,t): lane holds B[K = 4ks + 2*(lane>>4) + {0,1}][col = 16t + (lane&15)]
__global__ void k_pack_wref2(const float* __restrict__ Wref2, float2* __restrict__ bpack) {
  int idx = blockIdx.x * 256 + threadIdx.x;  // 64*16*32 = 32768
  int lane = idx & 31, t = (idx >> 5) & 15, ks = idx >> 9;
  int kk = 4 * ks + 2 * (lane >> 4);
  int col = 16 * t + (lane & 15);
  float2 v;
  v.x = Wref2[kk * D_ + col];
  v.y = Wref2[(kk + 1) * D_ + col];
  bpack[idx] = v;
}

// ---------------- 6. WMMA pointer head: u = 10*tanh(Vec2 . tanh(q2 + F@Wref2)) --------
// one wave per 16-row strip of F; full 16x256 f32 accumulator strip in registers.
__global__ __launch_bounds__(256) void k_pointer(const float* __restrict__ F,
                                                 const float* __restrict__ bpack,
                                                 const float* __restrict__ q2,
                                                 const float* __restrict__ vec2,
                                                 float* __restrict__ u) {
  int lane = threadIdx.x & 31;
  int wave = threadIdx.x >> 5;
  int strip = blockIdx.x * 8 + wave;  // 0..15999, exact
  int row0 = strip * 16;
  int b = row0 / N_;                  // strips never straddle batches (2000 = 125*16)
  int n0 = row0 - b * N_;
  int lo = lane & 15, hi = lane >> 4;

  v8f acc[16] = {};
  const float* arow = F + (long)(row0 + lo) * D_ + 2 * hi;
  const v2f* bp = (const v2f*)bpack;
  for (int ks = 0; ks < 64; ++ks) {
    v2f a = *(const v2f*)(arow + 4 * ks);
#pragma unroll
    for (int t = 0; t < 16; ++t) {
      v2f bf = bp[(ks * 16 + t) * 32 + lane];
      acc[t] = __builtin_amdgcn_wmma_f32_16x16x4_f32(
          false, a, false, bf, (short)0, acc[t], false, false);
    }
  }
  // epilogue: lane holds acc[t][i] = r2[row0 + i + 8*hi][16t + lo]
  float p[8];
#pragma unroll
  for (int i = 0; i < 8; ++i) p[i] = 0.f;
  for (int t = 0; t < 16; ++t) {
    int col = 16 * t + lo;
    float vv = vec2[col];
    float qv = q2[b * D_ + col];
#pragma unroll
    for (int i = 0; i < 8; ++i) p[i] += vv * tanhf(qv + acc[t][i]);
  }
#pragma unroll
  for (int m = 1; m < 16; m <<= 1)
#pragma unroll
    for (int i = 0; i < 8; ++i) p[i] += __shfl_xor(p[i], m, 32);
  if (lo == 0) {
#pragma unroll
    for (int i = 0; i < 8; ++i)
      u[(long)b * N_ + n0 + 8 * hi + i] = CLIP_ * tanhf(p[i]);
  }
}

// ---------------- 7. masked log_softmax over n per b ----------------
__global__ void k_logsoftmax(const float* __restrict__ u, const int* __restrict__ mask,
                             float* __restrict__ out) {
  int b = blockIdx.x, tid = threadIdx.x;
  __shared__ float red[256];
  float mx = -INFINITY;
  for (int n = tid; n < N_; n += 256) {
    float v = (mask[b * N_ + n] > 0) ? u[(long)b * N_ + n] : NEG_;
    mx = fmaxf(mx, v);
  }
  red[tid] = mx; __syncthreads();
  for (int s = 128; s > 0; s >>= 1) { if (tid < s) red[tid] = fmaxf(red[tid], red[tid + s]); __syncthreads(); }
  mx = red[0]; __syncthreads();
  float sm = 0.f;
  for (int n = tid; n < N_; n += 256) {
    float v = (mask[b * N_ + n] > 0) ? u[(long)b * N_ + n] : NEG_;
    sm += expf(v - mx);
  }
  red[tid] = sm; __syncthreads();
  for (int s = 128; s > 0; s >>= 1) { if (tid < s) red[tid] += red[tid + s]; __syncthreads(); }
  float lse = mx + logf(red[0]);
  for (int n = tid; n < N_; n += 256) {
    float v = (mask[b * N_ + n] > 0) ? u[(long)b * N_ + n] : NEG_;
    out[(long)b * N_ + n] = v - lse;
  }
}

extern "C" void kernel_launch(void* const* d_in, const int* in_sizes, int n_in,
                              void* d_out, int out_size, void* d_ws, size_t ws_size,
                              hipStream_t stream) {
  const float* F     = (const float*)d_in[0];
  const float* rs    = (const float*)d_in[1];
  const int*   mask  = (const int*)d_in[2];
  const float* E1    = (const float*)d_in[3];
  const float* b1    = (const float*)d_in[4];
  const float* E2    = (const float*)d_in[5];
  const float* b2    = (const float*)d_in[6];
  const float* E3    = (const float*)d_in[7];
  const float* b3    = (const float*)d_in[8];
  const float* Wq    = (const float*)d_in[9];
  const float* Wk    = (const float*)d_in[10];
  const float* Wv    = (const float*)d_in[11];
  const float* Wo    = (const float*)d_in[12];
  const float* Wq2   = (const float*)d_in[13];
  const float* Wref2 = (const float*)d_in[14];
  const float* Vec2  = (const float*)d_in[15];
  const float* v1    = (const float*)d_in[16];
  float* out = (float*)d_out;

  float* W = (float*)d_ws;
  float* query  = W;               // 32768
  float* qh     = W + 32768;       // 131072
  float* qk     = W + 163840;      // 131072
  float* Mp     = W + 294912;      // 8192   (B*CH*M)
  float* Sp     = W + 303104;      // 8192
  float* accp   = W + 311296;      // 2097152 (B*CH*M*D)
  float* fbar   = W + 2408448;     // 131072
  float* g      = W + 2539520;     // 131072
  float* query2 = W + 2670592;     // 32768
  float* q2     = W + 2703360;     // 32768
  float* ubuf   = W + 2736128;     // 256000
  float* bpack  = W + 2992128;     // 65536  (total ~12.2 MB)

  k_embed<<<1, 128, 0, stream>>>(rs, E1, b1, E2, b2, E3, b3, v1, query);
  k_qh<<<512, 256, 0, stream>>>(query, Wq, qh);
  k_qk<<<512, 256, 0, stream>>>(qh, Wk, qk);
  k_glimpse_flash<<<256, 256, 0, stream>>>(F, qk, mask, Mp, Sp, accp);
  k_glimpse_merge<<<B_ * M_, 256, 0, stream>>>(Mp, Sp, accp, fbar);
  k_g<<<512, 256, 0, stream>>>(fbar, Wv, g);
  k_query2<<<128, 256, 0, stream>>>(g, Wo, query2, out);
  k_q2<<<128, 256, 0, stream>>>(query2, Wq2, q2);
  k_pack_wref2<<<128, 256, 0, stream>>>(Wref2, (float2*)bpack);
  k_pointer<<<2000, 256, 0, stream>>>(F, bpack, q2, Vec2, ubuf);
  k_logsoftmax<<<B_, 256, 0, stream>>>(ubuf, mask, out);
}